// PhysioNetGeoLIF_4Class_22230750724617
// MI455X (gfx1250) — compile-verified
//
#include <hip/hip_runtime.h>
#include <math.h>

#define LEAK 0.9f
#define BN_EPS 1e-5f

typedef float v2f __attribute__((ext_vector_type(2)));
typedef float v8f __attribute__((ext_vector_type(8)));

constexpr int B = 256, T = 2048, C = 64, H = 64, NC = 4, DT = 50;
constexpr int M_TOTAL = B * T;               // 524288 rows of the GEMM
constexpr int NSTRIP  = M_TOTAL / 16;        // 32768 strips of 16 rows
constexpr int GEMM_BLOCKS  = 512;
constexpr int GEMM_THREADS = 256;            // 8 wave32 per block
constexpr int WAVES_PER_BLOCK = GEMM_THREADS / 32;

// workspace layout (float offsets)
constexpr size_t XP_OFF    = 0;                            // x_proj [M_TOTAL, H]
constexpr size_t PART_OFF  = (size_t)M_TOTAL * H;          // per-block stats [GEMM_BLOCKS][128]
constexpr size_t PART_SZ   = (size_t)GEMM_BLOCKS * 128;
constexpr size_t SCALE_OFF = PART_OFF + PART_SZ;           // [H]
constexpr size_t SHIFT_OFF = SCALE_OFF + H;                // [H]
constexpr size_t THRESH_OFF= SHIFT_OFF + H;                // [B,NC]
constexpr size_t WCL_OFF   = THRESH_OFF + (size_t)B * NC;  // [H,NC]

// ---------------------------------------------------------------------------
// K1: x_proj = X @ W^T via V_WMMA_F32_16X16X4_F32, plus per-channel sum/sumsq
// ---------------------------------------------------------------------------
__global__ __launch_bounds__(GEMM_THREADS)
void gemm_bn_stats_kernel(const float* __restrict__ X,
                          const float* __restrict__ Wsp,
                          float* __restrict__ xp,
                          float* __restrict__ partials)
{
    __shared__ float sW[H * C];     // 16 KB: W_spatial [H][C]
    __shared__ float sSum[H], sSS[H];

    const int tid = threadIdx.x;
    for (int i = tid; i < H * C; i += GEMM_THREADS) sW[i] = Wsp[i];
    if (tid < H) { sSum[tid] = 0.f; sSS[tid] = 0.f; }
    __syncthreads();

    const int lane  = tid & 31;
    const int l16   = lane & 15;
    const int hi    = lane >> 4;    // 0: K=0..1 half, 1: K=2..3 half
    const int gwave = blockIdx.x * WAVES_PER_BLOCK + (tid >> 5);
    const int nwave = gridDim.x * WAVES_PER_BLOCK;

    float s0 = 0.f, s1 = 0.f, s2 = 0.f, s3 = 0.f;
    float q0 = 0.f, q1 = 0.f, q2 = 0.f, q3 = 0.f;

    for (int strip = gwave; strip < NSTRIP; strip += nwave) {
        const int m0 = strip * 16;
        v8f acc0 = {0,0,0,0,0,0,0,0};
        v8f acc1 = {0,0,0,0,0,0,0,0};
        v8f acc2 = {0,0,0,0,0,0,0,0};
        v8f acc3 = {0,0,0,0,0,0,0,0};

        // A fragment source: row m0+l16, cols kk*4 + 2*hi (+0,+1)
        const float* arow = X + (size_t)(m0 + l16) * C + 2 * hi;

        #pragma unroll
        for (int kk = 0; kk < C / 4; ++kk) {
            const v2f a = *(const v2f*)(arow + kk * 4);
            // B fragment: B[c][h] = W[h][c] ; row h = nt*16+l16, col c = kk*4+2*hi
            const v2f b0 = *(const v2f*)(&sW[(0 * 16 + l16) * C + kk * 4 + 2 * hi]);
            const v2f b1 = *(const v2f*)(&sW[(1 * 16 + l16) * C + kk * 4 + 2 * hi]);
            const v2f b2 = *(const v2f*)(&sW[(2 * 16 + l16) * C + kk * 4 + 2 * hi]);
            const v2f b3 = *(const v2f*)(&sW[(3 * 16 + l16) * C + kk * 4 + 2 * hi]);
            acc0 = __builtin_amdgcn_wmma_f32_16x16x4_f32(false, a, false, b0, (short)0, acc0, false, false);
            acc1 = __builtin_amdgcn_wmma_f32_16x16x4_f32(false, a, false, b1, (short)0, acc1, false, false);
            acc2 = __builtin_amdgcn_wmma_f32_16x16x4_f32(false, a, false, b2, (short)0, acc2, false, false);
            acc3 = __builtin_amdgcn_wmma_f32_16x16x4_f32(false, a, false, b3, (short)0, acc3, false, false);
        }

        // D layout: VGPR v -> (M = v + 8*hi, N = l16) ; store + accumulate stats
        float* orow = xp + (size_t)m0 * H;
        #pragma unroll
        for (int v = 0; v < 8; ++v) {
            const int r = v + hi * 8;
            float* o = orow + (size_t)r * H + l16;
            const float e0 = acc0[v]; o[0]  = e0; s0 += e0; q0 += e0 * e0;
            const float e1 = acc1[v]; o[16] = e1; s1 += e1; q1 += e1 * e1;
            const float e2 = acc2[v]; o[32] = e2; s2 += e2; q2 += e2 * e2;
            const float e3 = acc3[v]; o[48] = e3; s3 += e3; q3 += e3 * e3;
        }
    }

    atomicAdd(&sSum[ 0 + l16], s0); atomicAdd(&sSS[ 0 + l16], q0);
    atomicAdd(&sSum[16 + l16], s1); atomicAdd(&sSS[16 + l16], q1);
    atomicAdd(&sSum[32 + l16], s2); atomicAdd(&sSS[32 + l16], q2);
    atomicAdd(&sSum[48 + l16], s3); atomicAdd(&sSS[48 + l16], q3);
    __syncthreads();
    if (tid < H) {
        partials[(size_t)blockIdx.x * 128 + tid]     = sSum[tid];
        partials[(size_t)blockIdx.x * 128 + 64 + tid] = sSS[tid];
    }
}

// ---------------------------------------------------------------------------
// K2: finalize BN scale/shift, dynamic thresholds, folded class matrix
// ---------------------------------------------------------------------------
__global__ __launch_bounds__(256)
void finalize_kernel(const float* __restrict__ partials,
                     const float* __restrict__ gamma, const float* __restrict__ beta,
                     const float* __restrict__ tda,   const float* __restrict__ Wtda,
                     const float* __restrict__ btda,  const float* __restrict__ Wclass,
                     const float* __restrict__ lateral,
                     float* __restrict__ scaleO, float* __restrict__ shiftO,
                     float* __restrict__ threshO, float* __restrict__ wclO)
{
    const int tid = threadIdx.x;

    // BN statistics -> scale/shift
    if (tid < H) {
        float S = 0.f, SS = 0.f;
        for (int blk = 0; blk < GEMM_BLOCKS; ++blk) {
            S  += partials[(size_t)blk * 128 + tid];
            SS += partials[(size_t)blk * 128 + 64 + tid];
        }
        const float invN = 1.0f / (float)M_TOTAL;
        const float mean = S * invN;
        const float var  = SS * invN - mean * mean;
        const float sc   = gamma[tid] * rsqrtf(var + BN_EPS);
        scaleO[tid] = sc;
        shiftO[tid] = beta[tid] - mean * sc;
    }

    // dynamic threshold per (b, c): 1 + 0.5*sigmoid(tda @ Wtda^T + btda)
    {
        const int b = tid;  // B == 256 == blockDim
        float a0 = btda[0], a1 = btda[1], a2 = btda[2], a3 = btda[3];
        for (int j = 0; j < DT; ++j) {
            const float x = tda[(size_t)b * DT + j];
            a0 += x * Wtda[0 * DT + j];
            a1 += x * Wtda[1 * DT + j];
            a2 += x * Wtda[2 * DT + j];
            a3 += x * Wtda[3 * DT + j];
        }
        threshO[b * 4 + 0] = 1.0f + 0.5f / (1.0f + expf(-a0));
        threshO[b * 4 + 1] = 1.0f + 0.5f / (1.0f + expf(-a1));
        threshO[b * 4 + 2] = 1.0f + 0.5f / (1.0f + expf(-a2));
        threshO[b * 4 + 3] = 1.0f + 0.5f / (1.0f + expf(-a3));
    }

    // Wcl[h][c] = sum_k Wclass[k][h] * lateral[k][c]   (256 entries == blockDim)
    {
        const int h = tid >> 2, c = tid & 3;
        float w = 0.f;
        for (int k = 0; k < NC; ++k) w += Wclass[k * H + h] * lateral[k * NC + c];
        wclO[tid] = w;
    }
}

// ---------------------------------------------------------------------------
// K3: LIF scan — one wave32 per batch element, 2 channels per lane
// ---------------------------------------------------------------------------
__global__ __launch_bounds__(32)
void lif_scan_kernel(const float* __restrict__ xp,
                     const float* __restrict__ scale, const float* __restrict__ shift,
                     const float* __restrict__ thresh, const float* __restrict__ wcl,
                     float* __restrict__ out)
{
    const int b    = blockIdx.x;
    const int lane = threadIdx.x;
    const int h0   = 2 * lane, h1 = 2 * lane + 1;

    const float  sc0 = scale[h0], sc1 = scale[h1];
    const float  sh0 = shift[h0], sh1 = shift[h1];
    const float4 w0  = *(const float4*)(wcl + h0 * 4);
    const float4 w1  = *(const float4*)(wcl + h1 * 4);
    const float4 th  = *(const float4*)(thresh + b * 4);

    float  m1a = 0.f, m1b = 0.f;
    float4 m2   = make_float4(0.f, 0.f, 0.f, 0.f);
    float4 ssum = make_float4(0.f, 0.f, 0.f, 0.f);

    const float2* xr = (const float2*)(xp + (size_t)b * T * H);
    float2 xv = xr[lane];                       // t = 0, h = 2*lane, 2*lane+1

    for (int t = 0; t < T; ++t) {
        // prefetch next row (256B/wave, L2-resident) behind the dependent chain
        float2 xn = (t + 1 < T) ? xr[(size_t)(t + 1) * (H / 2) + lane]
                                : make_float2(0.f, 0.f);

        // layer 1 LIF (batchnorm folded into scale/shift)
        m1a = LEAK * m1a + fmaf(xv.x, sc0, sh0);
        m1b = LEAK * m1b + fmaf(xv.y, sc1, sh1);
        const float sa = (m1a >= 1.0f) ? 1.0f : 0.0f;
        const float sb = (m1b >= 1.0f) ? 1.0f : 0.0f;
        m1a = (sa > 0.f) ? 0.f : m1a;
        m1b = (sb > 0.f) ? 0.f : m1b;

        // class currents: spk1 @ (Wclass^T @ lateral), butterfly-reduced
        float px = sa * w0.x + sb * w1.x;
        float py = sa * w0.y + sb * w1.y;
        float pz = sa * w0.z + sb * w1.z;
        float pw = sa * w0.w + sb * w1.w;
        #pragma unroll
        for (int off = 16; off >= 1; off >>= 1) {
            px += __shfl_xor(px, off, 32);
            py += __shfl_xor(py, off, 32);
            pz += __shfl_xor(pz, off, 32);
            pw += __shfl_xor(pw, off, 32);
        }

        // layer 2 LIF with dynamic threshold
        m2.x = LEAK * m2.x + px;
        m2.y = LEAK * m2.y + py;
        m2.z = LEAK * m2.z + pz;
        m2.w = LEAK * m2.w + pw;
        const float t0 = (m2.x >= th.x) ? 1.f : 0.f;
        const float t1 = (m2.y >= th.y) ? 1.f : 0.f;
        const float t2 = (m2.z >= th.z) ? 1.f : 0.f;
        const float t3 = (m2.w >= th.w) ? 1.f : 0.f;
        ssum.x += t0; ssum.y += t1; ssum.z += t2; ssum.w += t3;
        m2.x = (t0 > 0.f) ? 0.f : m2.x;
        m2.y = (t1 > 0.f) ? 0.f : m2.y;
        m2.z = (t2 > 0.f) ? 0.f : m2.z;
        m2.w = (t3 > 0.f) ? 0.f : m2.w;

        xv = xn;
    }

    if (lane == 0) *(float4*)(out + (size_t)b * 4) = ssum;
}

// ---------------------------------------------------------------------------
extern "C" void kernel_launch(void* const* d_in, const int* in_sizes, int n_in,
                              void* d_out, int out_size, void* d_ws, size_t ws_size,
                              hipStream_t stream)
{
    (void)in_sizes; (void)n_in; (void)out_size; (void)ws_size;
    const float* X       = (const float*)d_in[0];   // [B,T,C]
    const float* tda     = (const float*)d_in[1];   // [B,DT]
    const float* Wsp     = (const float*)d_in[2];   // [H,C]
    const float* gamma   = (const float*)d_in[3];   // [H]
    const float* beta    = (const float*)d_in[4];   // [H]
    const float* Wclass  = (const float*)d_in[5];   // [NC,H]
    const float* lateral = (const float*)d_in[6];   // [NC,NC]
    const float* Wtda    = (const float*)d_in[7];   // [NC,DT]
    const float* btda    = (const float*)d_in[8];   // [NC]

    float* ws       = (float*)d_ws;
    float* xp       = ws + XP_OFF;
    float* partials = ws + PART_OFF;
    float* scale    = ws + SCALE_OFF;
    float* shift    = ws + SHIFT_OFF;
    float* thresh   = ws + THRESH_OFF;
    float* wcl      = ws + WCL_OFF;

    gemm_bn_stats_kernel<<<GEMM_BLOCKS, GEMM_THREADS, 0, stream>>>(X, Wsp, xp, partials);
    finalize_kernel<<<1, 256, 0, stream>>>(partials, gamma, beta, tda, Wtda, btda,
                                           Wclass, lateral, scale, shift, thresh, wcl);
    lif_scan_kernel<<<B, 32, 0, stream>>>(xp, scale, shift, thresh, wcl, (float*)d_out);
}